// PointNetSetAbstraction_16930761080948
// MI455X (gfx1250) — compile-verified
//
#include <hip/hip_runtime.h>
#include <math.h>

// PointNet++ Set Abstraction for MI455X (gfx1250, wave32).
// FPS -> radius group/gather -> 3x (WMMA f32 GEMM, BN fused) -> masked maxpool.
// GEMMs: V_WMMA_F32_16X16X4_F32; K padded at compile time (guard-free loop);
// W tile + BN coeffs staged in LDS; B streamed from HBM with prefetch.
// BN statistics are produced in the GEMM epilogue (deterministic per-tile
// partials, no atomics); BN apply+ReLU is fused into the *consumer* of each
// layer (next GEMM's B-load, or the final maxpool), eliminating all
// standalone stats/apply passes over the 67-134 MB activation tensors.

#define BATCH 8
#define NPTS  4096
#define CFEAT 64
#define MCENT 1024
#define KNBR  32
#define PTOT  (BATCH * MCENT * KNBR)   // 262144 = 2^18 positions
#define NTILE (PTOT / 16)              // 16384 N-tiles per channel
#define RAD2  0.04f
#define BNEPS 1e-5f

typedef __attribute__((ext_vector_type(2))) float v2f;
typedef __attribute__((ext_vector_type(8))) float v8f;

// ---------------------------------------------------------------------------
// wave32 reductions (all lanes active; deterministic)
// ---------------------------------------------------------------------------
__device__ __forceinline__ void wave_argmax(float& v, int& i) {
#pragma unroll
    for (int off = 16; off > 0; off >>= 1) {
        float ov = __shfl_down(v, off, 32);
        int   oi = __shfl_down(i, off, 32);
        if (ov > v || (ov == v && oi < i)) { v = ov; i = oi; }
    }
}
__device__ __forceinline__ void wave_argmin(float& v, int& i) {
#pragma unroll
    for (int off = 16; off > 0; off >>= 1) {
        float ov = __shfl_down(v, off, 32);
        int   oi = __shfl_down(i, off, 32);
        if (ov < v || (ov == v && oi < i)) { v = ov; i = oi; }
    }
}
__device__ __forceinline__ void wave_sum2(float& s, float& q) {
#pragma unroll
    for (int off = 16; off > 0; off >>= 1) {
        s += __shfl_down(s, off, 32);
        q += __shfl_down(q, off, 32);
    }
}

// ---------------------------------------------------------------------------
// 1) Farthest point sampling: one block per batch. Points + running distance
//    in registers (16 pts/thread); cloud staged in LDS for the uniform
//    centroid fetch on the 1024-step serial chain. Matches jnp.argmax
//    first-occurrence tie-break.
// ---------------------------------------------------------------------------
__global__ __launch_bounds__(256) void fps_kernel(const float* __restrict__ xyz,
                                                  int* __restrict__ idx_out,
                                                  float* __restrict__ newxyz_out)
{
    __shared__ float sx[NPTS], sy[NPTS], sz[NPTS];
    __shared__ float wv[8];
    __shared__ int   wi[8];
    __shared__ int   sfar;
    const int b = blockIdx.x, t = threadIdx.x;
    const int lane = t & 31, wave = t >> 5;
    const float* px = xyz + (size_t)b * NPTS * 3;

    float X[16], Yc[16], Z[16], D[16];
#pragma unroll
    for (int s = 0; s < 16; ++s) {
        const int i = t + s * 256;
        float x = px[i * 3 + 0], y = px[i * 3 + 1], z = px[i * 3 + 2];
        X[s] = x; Yc[s] = y; Z[s] = z; D[s] = 1e10f;
        sx[i] = x; sy[i] = y; sz[i] = z;
    }
    int far = 0;
    __syncthreads();

    for (int it = 0; it < MCENT; ++it) {
        const float cx = sx[far], cy = sy[far], cz = sz[far];
        if (t == 0) {
            idx_out[b * MCENT + it] = far;
            newxyz_out[(b * MCENT + it) * 3 + 0] = cx;
            newxyz_out[(b * MCENT + it) * 3 + 1] = cy;
            newxyz_out[(b * MCENT + it) * 3 + 2] = cz;
        }
        float bv = -1.0f; int bs = 0;
#pragma unroll
        for (int s = 0; s < 16; ++s) {
            float dx = X[s] - cx, dy = Yc[s] - cy, dz = Z[s] - cz;
            float d  = dx * dx + dy * dy + dz * dz;
            float nd = fminf(D[s], d);
            D[s] = nd;
            if (nd > bv) { bv = nd; bs = s; }      // strict > keeps lowest slot
        }
        int gi = bs * 256 + t;
        wave_argmax(bv, gi);
        if (lane == 0) { wv[wave] = bv; wi[wave] = gi; }
        __syncthreads();
        if (t == 0) {
            float fb = wv[0]; int fi = wi[0];
#pragma unroll
            for (int w = 1; w < 8; ++w)
                if (wv[w] > fb || (wv[w] == fb && wi[w] < fi)) { fb = wv[w]; fi = wi[w]; }
            sfar = fi;
        }
        __syncthreads();
        far = sfar;
    }
}

// ---------------------------------------------------------------------------
// 2) Radius grouping + gather: one block per centroid, d2 in registers.
//    32 iterative argmins (set == stable argsort top-K), then cooperative
//    gather into channel-major, K-padded X0 (68 x PTOT).
// ---------------------------------------------------------------------------
__global__ __launch_bounds__(256) void group_kernel(const float* __restrict__ xyz,
                                                    const float* __restrict__ feats,
                                                    const float* __restrict__ newxyz,
                                                    int* __restrict__ mask_out,
                                                    float* __restrict__ X0)
{
    __shared__ float wv[8];
    __shared__ int   wi[8];
    __shared__ int   sel[KNBR];
    __shared__ int   smk[KNBR];
    const int bm = blockIdx.x;              // b*M + m
    const int b  = bm >> 10;
    const int t  = threadIdx.x;
    const int lane = t & 31, wave = t >> 5;
    const float cx = newxyz[bm * 3 + 0];
    const float cy = newxyz[bm * 3 + 1];
    const float cz = newxyz[bm * 3 + 2];
    const float c2 = cx * cx + cy * cy + cz * cz;
    const float* px = xyz + (size_t)b * NPTS * 3;

    float d2r[16];
#pragma unroll
    for (int s = 0; s < 16; ++s) {
        const int i = t + s * 256;
        float x = px[i * 3 + 0], y = px[i * 3 + 1], z = px[i * 3 + 2];
        d2r[s] = c2 + (x * x + y * y + z * z) - 2.0f * (cx * x + cy * y + cz * z);
    }

    for (int k = 0; k < KNBR; ++k) {
        float bv = 3.4e38f; int bs = 0;
#pragma unroll
        for (int s = 0; s < 16; ++s)
            if (d2r[s] < bv) { bv = d2r[s]; bs = s; }
        int gi = bs * 256 + t;
        wave_argmin(bv, gi);
        if (lane == 0) { wv[wave] = bv; wi[wave] = gi; }
        __syncthreads();
        if (t == 0) {
            float fb = wv[0]; int fi = wi[0];
#pragma unroll
            for (int w = 1; w < 8; ++w)
                if (wv[w] < fb || (wv[w] == fb && wi[w] < fi)) { fb = wv[w]; fi = wi[w]; }
            sel[k] = fi;
            smk[k] = (fb <= RAD2) ? 1 : 0;
        }
        __syncthreads();
        const int s    = sel[k];
        const bool own = (s & 255) == t;
        const int slot = s >> 8;
#pragma unroll
        for (int q = 0; q < 16; ++q)          // constant-index invalidation (no scratch)
            if (own && slot == q) d2r[q] = 3.4e38f;
    }

    const int pbase = bm * KNBR;
    for (int e = t; e < KNBR * 68; e += 256) {
        int k = e / 68, c = e - k * 68;
        int n = sel[k];
        float val;
        if (c < CFEAT)      val = feats[((size_t)b * CFEAT + c) * NPTS + n];
        else if (c < 67) {
            int d = c - CFEAT;
            float pv = px[n * 3 + d];
            float cv = (d == 0) ? cx : ((d == 1) ? cy : cz);
            val = pv - cv;                    // grouped_xyz_norm
        } else              val = 0.0f;       // K-pad row 67
        X0[(size_t)c * PTOT + pbase + k] = val;
    }
    for (int k = t; k < KNBR; k += 256) mask_out[pbase + k] = smk[k];
}

// ---------------------------------------------------------------------------
// 3a) Pad w0 (64x67) -> (64x68) with zeros so the GEMM K-loop is guard-free.
// ---------------------------------------------------------------------------
__global__ __launch_bounds__(256) void padw_kernel(const float* __restrict__ w,
                                                   float* __restrict__ wp,
                                                   int Cout, int Cin, int CinPad)
{
    int i = blockIdx.x * 256 + threadIdx.x;
    if (i >= Cout * CinPad) return;
    int m = i / CinPad, k = i - m * CinPad;
    wp[i] = (k < Cin) ? w[m * Cin + k] : 0.0f;
}

// ---------------------------------------------------------------------------
// 3b) GEMM via V_WMMA_F32_16X16X4_F32, one 16x16 tile per wave, 8 waves/block.
//     FUSEIN: apply previous layer's BN (relu(x*sc+sh)) to B-fragments
//     in-register (coeffs staged in LDS). Epilogue: store raw Y and emit
//     deterministic per-(channel, N-tile) BN partial sums (no atomics).
// ---------------------------------------------------------------------------
template<int CINP, bool FUSEIN>
__global__ __launch_bounds__(256) void gemm_wmma(const float* __restrict__ Wp,
                                                 const float* __restrict__ X,
                                                 float* __restrict__ Y,
                                                 const float* __restrict__ isc,
                                                 const float* __restrict__ ish,
                                                 float* __restrict__ psum,
                                                 float* __restrict__ psq)
{
    __shared__ float lw[16 * CINP];
    __shared__ float lsc[CINP], lsh[CINP];
    const int t  = threadIdx.x;
    const int m0 = blockIdx.y * 16;
    for (int e = t; e < 16 * CINP; e += 256)          // rows m0..m0+15 contiguous
        lw[e] = Wp[(size_t)m0 * CINP + e];
    if (FUSEIN)
        for (int e = t; e < CINP; e += 256) { lsc[e] = isc[e]; lsh[e] = ish[e]; }
    __syncthreads();

    const int lane = t & 31;
    const int wave = t >> 5;
    const int row  = lane & 15;
    const int hi   = lane >> 4;                        // half-wave: 0 or 1
    const int n    = (blockIdx.x * 8 + wave) * 16 + row;
    const float* __restrict__ xp = X + n;

    v8f acc = {};
#pragma unroll 4
    for (int k0 = 0; k0 < CINP; k0 += 4) {
        const int ka = k0 + 2 * hi;                    // even -> 8B-aligned LDS reads
        v2f a = *(const v2f*)&lw[row * CINP + ka];
        v2f bfrag;
        bfrag.x = xp[(size_t)ka * PTOT];
        bfrag.y = xp[(size_t)(ka + 1) * PTOT];
        if (FUSEIN) {                                  // fused BN+ReLU of producer
            v2f sc = *(const v2f*)&lsc[ka];
            v2f sh = *(const v2f*)&lsh[ka];
            bfrag.x = fmaxf(0.0f, bfrag.x * sc.x + sh.x);
            bfrag.y = fmaxf(0.0f, bfrag.y * sc.y + sh.y);
        }
        if (k0 + 8 < CINP)                             // uniform guard
            __builtin_prefetch(&xp[(size_t)(ka + 8) * PTOT], 0, 1);
        acc = __builtin_amdgcn_wmma_f32_16x16x4_f32(
            /*neg_a=*/false, a, /*neg_b=*/false, bfrag,
            /*c_mod=*/(short)0, acc, /*reuse_a=*/false, /*reuse_b=*/false);
    }

    const int nb = blockIdx.x * 8 + wave;              // N-tile index (0..NTILE-1)
#pragma unroll
    for (int v = 0; v < 8; ++v) {
        const int m = m0 + v + 8 * hi;                 // channel of this C/D row
        float val = acc[v];
        Y[(size_t)m * PTOT + n] = val;                 // raw (pre-BN) activation
        float s = val, q = val * val;                  // row-wise BN partials
#pragma unroll
        for (int off = 8; off > 0; off >>= 1) {        // reduce the 16 cols
            s += __shfl_down(s, off, 16);
            q += __shfl_down(q, off, 16);
        }
        if (row == 0) {                                // lanes 0 and 16
            psum[(size_t)m * NTILE + nb] = s;
            psq [(size_t)m * NTILE + nb] = q;
        }
    }
}

// ---------------------------------------------------------------------------
// 4) Fold per-tile partials -> per-channel scale/shift. One block per channel,
//    deterministic (fixed partition + wave shfl + fixed cross-wave order).
// ---------------------------------------------------------------------------
__global__ __launch_bounds__(256) void bnred_kernel(const float* __restrict__ psum,
                                                    const float* __restrict__ psq,
                                                    const float* __restrict__ g,
                                                    const float* __restrict__ bb,
                                                    float* __restrict__ scale,
                                                    float* __restrict__ shift)
{
    __shared__ float ws_[8], wq[8];
    const int c = blockIdx.x;
    const int t = threadIdx.x, lane = t & 31, wave = t >> 5;
    const float4* ps = (const float4*)(psum + (size_t)c * NTILE);
    const float4* pq = (const float4*)(psq  + (size_t)c * NTILE);
    float s = 0.0f, q = 0.0f;
    for (int i = t; i < NTILE / 4; i += 256) {
        float4 a = ps[i]; s += a.x + a.y + a.z + a.w;
        float4 d = pq[i]; q += d.x + d.y + d.z + d.w;
    }
    wave_sum2(s, q);
    if (lane == 0) { ws_[wave] = s; wq[wave] = q; }
    __syncthreads();
    if (t == 0) {
        float fs = 0.0f, fq = 0.0f;
#pragma unroll
        for (int w = 0; w < 8; ++w) { fs += ws_[w]; fq += wq[w]; }
        float mean = fs / (float)PTOT;
        float var  = fq / (float)PTOT - mean * mean;
        float inv  = rsqrtf(var + BNEPS);
        float sc   = g[c] * inv;
        scale[c] = sc;
        shift[c] = bb[c] - mean * sc;
    }
}

// ---------------------------------------------------------------------------
// 5) Masked max-pool over K neighbors with layer-2 BN+ReLU fused in.
// ---------------------------------------------------------------------------
__global__ __launch_bounds__(128) void maxpool_kernel(const float* __restrict__ Y2,
                                                      const int* __restrict__ mask,
                                                      const float* __restrict__ scale,
                                                      const float* __restrict__ shift,
                                                      float* __restrict__ out)
{
    __shared__ int smk[KNBR];
    const int bm    = blockIdx.x;
    const int co    = threadIdx.x;                // 0..127
    const int pbase = bm * KNBR;
    if (co < KNBR) smk[co] = mask[pbase + co];
    __syncthreads();
    const float sc = scale[co], sh = shift[co];
    const float4* yp = (const float4*)(Y2 + (size_t)co * PTOT + pbase);
    float mx = -INFINITY;
#pragma unroll
    for (int k = 0; k < KNBR / 4; ++k) {
        float4 v = yp[k];
        float y0 = fmaxf(0.0f, v.x * sc + sh);
        float y1 = fmaxf(0.0f, v.y * sc + sh);
        float y2 = fmaxf(0.0f, v.z * sc + sh);
        float y3 = fmaxf(0.0f, v.w * sc + sh);
        if (smk[4 * k + 0]) mx = fmaxf(mx, y0);
        if (smk[4 * k + 1]) mx = fmaxf(mx, y1);
        if (smk[4 * k + 2]) mx = fmaxf(mx, y2);
        if (smk[4 * k + 3]) mx = fmaxf(mx, y3);
    }
    const int b = bm >> 10, m = bm & (MCENT - 1);
    out[((size_t)b * 128 + co) * MCENT + m] = mx;
}

// ---------------------------------------------------------------------------
extern "C" void kernel_launch(void* const* d_in, const int* in_sizes, int n_in,
                              void* d_out, int out_size, void* d_ws, size_t ws_size,
                              hipStream_t stream)
{
    const float* xyz   = (const float*)d_in[0];
    const float* feats = (const float*)d_in[1];
    const float* w0 = (const float*)d_in[2];
    const float* g0 = (const float*)d_in[3];
    const float* b0 = (const float*)d_in[4];
    const float* w1 = (const float*)d_in[5];
    const float* g1 = (const float*)d_in[6];
    const float* b1 = (const float*)d_in[7];
    const float* w2 = (const float*)d_in[8];
    const float* g2 = (const float*)d_in[9];
    const float* b2 = (const float*)d_in[10];

    float* out_xyz  = (float*)d_out;                              // (B,M,3)
    float* out_feat = (float*)d_out + (size_t)BATCH * MCENT * 3;  // (B,128,M)

    // Workspace layout (element offsets):
    int*   wsi     = (int*)d_ws;
    float* wsf     = (float*)d_ws;
    int*   fps_idx = wsi;                         // 8192 ints
    int*   mask    = wsi + 8192;                  // PTOT ints
    float* scale   = wsf + 8192 + PTOT;           // 128
    float* shift   = scale + 128;                 // 128
    float* w0pad   = shift + 128;                 // 64*68 = 4352
    size_t off = (size_t)(8192 + PTOT + 128 + 128 + 4352);
    off = (off + 255) & ~(size_t)255;
    float* psum = wsf + off;                      // 128*NTILE = 2M floats
    float* psq  = psum + (size_t)128 * NTILE;     // 2M floats
    size_t off2 = off + (size_t)256 * NTILE;
    float* bufA = wsf + off2;                     // 68*PTOT: X0, later Y1/X2
    float* bufB = bufA + (size_t)68 * PTOT;       // 64*PTOT: Y0/X1
    float* bufC = bufB + (size_t)64 * PTOT;       // 128*PTOT: Y2

    fps_kernel<<<BATCH, 256, 0, stream>>>(xyz, fps_idx, out_xyz);
    group_kernel<<<BATCH * MCENT, 256, 0, stream>>>(xyz, feats, out_xyz, mask, bufA);
    padw_kernel<<<(64 * 68 + 255) / 256, 256, 0, stream>>>(w0, w0pad, 64, 67, 68);

    // layer 0: raw Y0(64xP) = w0pad(64x68) * X0(68xP); emit BN0 partials
    gemm_wmma<68, false><<<dim3(PTOT / 128, 4), 256, 0, stream>>>(
        w0pad, bufA, bufB, scale, shift, psum, psq);
    bnred_kernel<<<64, 256, 0, stream>>>(psum, psq, g0, b0, scale, shift);

    // layer 1: raw Y1 = w1 * relu(bn0(Y0)); emit BN1 partials
    gemm_wmma<64, true><<<dim3(PTOT / 128, 4), 256, 0, stream>>>(
        w1, bufB, bufA, scale, shift, psum, psq);
    bnred_kernel<<<64, 256, 0, stream>>>(psum, psq, g1, b1, scale, shift);

    // layer 2: raw Y2 = w2 * relu(bn1(Y1)); emit BN2 partials
    gemm_wmma<64, true><<<dim3(PTOT / 128, 8), 256, 0, stream>>>(
        w2, bufA, bufC, scale, shift, psum, psq);
    bnred_kernel<<<128, 256, 0, stream>>>(psum, psq, g2, b2, scale, shift);

    // masked max-pool with BN2+ReLU fused
    maxpool_kernel<<<BATCH * MCENT, 128, 0, stream>>>(bufC, mask, scale, shift, out_feat);
}